// Conv_layer_46875273068969
// MI455X (gfx1250) — compile-verified
//
#include <hip/hip_runtime.h>

#define D 128
#define PITCH 132   // 128 + 4 pad: row stride 132 mod 64 banks = 4 -> conflict-free column access
#define NEG_SLOPE 0.01f

typedef __attribute__((ext_vector_type(2))) float v2f;
typedef __attribute__((ext_vector_type(8))) float v8f;

// ---------------------------------------------------------------------------
// Zero a float buffer (workspace agg must be zeroed every call; the harness
// poisons and never re-initializes).
// ---------------------------------------------------------------------------
__global__ void zero_f32(float* __restrict__ p, long n) {
    long i = (long)blockIdx.x * blockDim.x + threadIdx.x;
    if (i < n) p[i] = 0.0f;
}

// ---------------------------------------------------------------------------
// Edge scatter: agg[dst[e], :] += feat[src[e], :] * w[e]
// One wave32 per edge; each lane handles a contiguous float4 (32*4 = 128 = D).
// Gather is a coalesced 512B row read; atomics land in a 25.6MB buffer that
// fits in MI455X's 192MB L2, so the RMW traffic stays on-chip.
// ---------------------------------------------------------------------------
__global__ void scatter_edges(const float* __restrict__ feat,
                              const int*   __restrict__ src,
                              const int*   __restrict__ dst,
                              const float* __restrict__ w,
                              float*       __restrict__ agg,
                              int ne) {
    int t    = blockIdx.x * blockDim.x + threadIdx.x;
    int e    = t >> 5;
    int lane = t & 31;
    if (e >= ne) return;
    int   s  = src[e];
    int   d2 = dst[e];
    float ww = w[e];
    float4 v = ((const float4*)(feat + (long)s * D))[lane];
    float* o = agg + (long)d2 * D + lane * 4;
    atomicAdd(o + 0, v.x * ww);
    atomicAdd(o + 1, v.y * ww);
    atomicAdd(o + 2, v.z * ww);
    atomicAdd(o + 3, v.w * ww);
}

// ---------------------------------------------------------------------------
// GIN MLP: out = leaky(leaky(X @ W1 + b1) @ W2 + b2) + feat   (residual)
//          X   = (1+eps)*feat + agg
// Optionally saves the pre-residual MLP result (needed as message source for
// the second conv direction).
//
// Block = 256 threads = 8 waves; block owns 16 rows (M-tile), wave j owns the
// 16-column N-slab [16j, 16j+16). fp32 WMMA 16x16x4, K-loop of 32 steps.
//
// ISA fragment layouts (CDNA5 ISA):
//   A 16x4 : lane L -> M = L&15,  v0 = K(k0 + 2*(L>>4)),     v1 = K(+1)
//   B 4x16 : lane L -> N = L&15,  v0 = K(k0 + 2*(L>>4)),     v1 = K(+1)
//   C 16x16: VGPR g, lane L -> M = g + 8*(L>>4), N = L&15
// ---------------------------------------------------------------------------
__global__ __launch_bounds__(256)
void gin_mlp(const float* __restrict__ feat,   // [N, D] pre-conv features
             const float* __restrict__ agg,    // [N, D] aggregated messages
             const float* __restrict__ W1,     // [D, D] row-major (k, n)
             const float* __restrict__ b1,     // [D]
             const float* __restrict__ W2,     // [D, D]
             const float* __restrict__ b2,     // [D]
             const float* __restrict__ epsp,   // scalar
             float*       __restrict__ out,    // [N, D] mlp + feat
             float*       __restrict__ hsave)  // [N, D] mlp (pre-residual) or null
{
    __shared__ float lds_x[16 * PITCH];
    __shared__ float lds_h[16 * PITCH];

    const int row0 = blockIdx.x << 4;   // 16 rows per block; 50000 = 3125*16 exactly
    const float eps1 = 1.0f + *epsp;

    // ---- stage X = (1+eps)*feat + agg into LDS (each thread: 1 row, 8 cols)
    {
        const int tr = threadIdx.x >> 4;          // 0..15
        const int tc = (threadIdx.x & 15) << 3;   // 0,8,...,120
        const float4* fr = (const float4*)(feat + (long)(row0 + tr) * D + tc);
        const float4* ar = (const float4*)(agg  + (long)(row0 + tr) * D + tc);
        float4 f0 = fr[0], f1 = fr[1];
        float4 a0 = ar[0], a1 = ar[1];
        float* xr = &lds_x[tr * PITCH + tc];
        xr[0] = eps1 * f0.x + a0.x;  xr[1] = eps1 * f0.y + a0.y;
        xr[2] = eps1 * f0.z + a0.z;  xr[3] = eps1 * f0.w + a0.w;
        xr[4] = eps1 * f1.x + a1.x;  xr[5] = eps1 * f1.y + a1.y;
        xr[6] = eps1 * f1.z + a1.z;  xr[7] = eps1 * f1.w + a1.w;
    }
    __syncthreads();

    const int lane  = threadIdx.x & 31;
    const int wave  = threadIdx.x >> 5;     // 0..7
    const int nbase = wave << 4;            // N-slab base
    const int mn    = lane & 15;            // M (for A) / N (for B,D)
    const int koff  = (lane >> 4) << 1;     // K sub-offset: 0 or 2
    const int rbase = (lane >> 4) << 3;     // D-row base: 0 or 8

    // ---- layer 1: H = leaky(X @ W1 + b1), H -> LDS
    v8f acc = {0.f, 0.f, 0.f, 0.f, 0.f, 0.f, 0.f, 0.f};
#pragma unroll
    for (int k0 = 0; k0 < D; k0 += 4) {
        v2f a = *(const v2f*)&lds_x[mn * PITCH + k0 + koff];
        v2f b;
        b.x = W1[(k0 + koff    ) * D + nbase + mn];
        b.y = W1[(k0 + koff + 1) * D + nbase + mn];
        acc = __builtin_amdgcn_wmma_f32_16x16x4_f32(false, a, false, b,
                                                    (short)0, acc, false, false);
    }
    {
        const float bs = b1[nbase + mn];
#pragma unroll
        for (int g = 0; g < 8; ++g) {
            float v = acc[g] + bs;
            v = (v >= 0.0f) ? v : NEG_SLOPE * v;
            lds_h[(rbase + g) * PITCH + nbase + mn] = v;
        }
    }
    __syncthreads();

    // ---- layer 2: Y = leaky(H @ W2 + b2); out = Y + feat; hsave = Y
    v8f acc2 = {0.f, 0.f, 0.f, 0.f, 0.f, 0.f, 0.f, 0.f};
#pragma unroll
    for (int k0 = 0; k0 < D; k0 += 4) {
        v2f a = *(const v2f*)&lds_h[mn * PITCH + k0 + koff];
        v2f b;
        b.x = W2[(k0 + koff    ) * D + nbase + mn];
        b.y = W2[(k0 + koff + 1) * D + nbase + mn];
        acc2 = __builtin_amdgcn_wmma_f32_16x16x4_f32(false, a, false, b,
                                                     (short)0, acc2, false, false);
    }
    {
        const float bs = b2[nbase + mn];
#pragma unroll
        for (int g = 0; g < 8; ++g) {
            float v = acc2[g] + bs;
            v = (v >= 0.0f) ? v : NEG_SLOPE * v;
            const long idx = (long)(row0 + rbase + g) * D + nbase + mn;
            out[idx] = v + feat[idx];
            if (hsave) hsave[idx] = v;
        }
    }
}

// ---------------------------------------------------------------------------
// Host-side launch (graph-capture safe: kernels only, all on `stream`).
// d_ws layout: [agg: max(N_ROW,N_COL)*D f32][h_row: N_ROW*D f32]  (~51.2 MB)
// ---------------------------------------------------------------------------
extern "C" void kernel_launch(void* const* d_in, const int* in_sizes, int n_in,
                              void* d_out, int out_size, void* d_ws, size_t ws_size,
                              hipStream_t stream) {
    const float* feat_row = (const float*)d_in[0];
    const float* feat_col = (const float*)d_in[1];
    const int*   src_c2r  = (const int*)  d_in[2];
    const int*   dst_c2r  = (const int*)  d_in[3];
    const float* w_c2r    = (const float*)d_in[4];
    const int*   src_r2c  = (const int*)  d_in[5];
    const int*   dst_r2c  = (const int*)  d_in[6];
    const float* w_r2c    = (const float*)d_in[7];
    const float* W1_c2r   = (const float*)d_in[8];
    const float* b1_c2r   = (const float*)d_in[9];
    const float* W2_c2r   = (const float*)d_in[10];
    const float* b2_c2r   = (const float*)d_in[11];
    const float* W1_r2c   = (const float*)d_in[12];
    const float* b1_r2c   = (const float*)d_in[13];
    const float* W2_r2c   = (const float*)d_in[14];
    const float* b2_r2c   = (const float*)d_in[15];
    const float* eps_c2r  = (const float*)d_in[16];
    const float* eps_r2c  = (const float*)d_in[17];

    const long nrow = in_sizes[0] / D;   // 50000
    const long ncol = in_sizes[1] / D;   // 50000
    const int  ne   = in_sizes[2];       // 600000
    const long nmax = (nrow > ncol) ? nrow : ncol;

    float* agg   = (float*)d_ws;
    float* h_row = agg + (size_t)nmax * D;

    float* out_row = (float*)d_out;
    float* out_col = out_row + (size_t)nrow * D;

    const int sblk = (int)(((long)ne * 32 + 255) / 256);

    // ---- col -> row GINConv
    {
        long na = nrow * (long)D;
        zero_f32<<<(int)((na + 255) / 256), 256, 0, stream>>>(agg, na);
        scatter_edges<<<sblk, 256, 0, stream>>>(feat_col, src_c2r, dst_c2r, w_c2r, agg, ne);
        gin_mlp<<<(int)(nrow / 16), 256, 0, stream>>>(
            feat_row, agg, W1_c2r, b1_c2r, W2_c2r, b2_c2r, eps_c2r, out_row, h_row);
    }
    // ---- row -> col GINConv (messages use updated h_row)
    {
        long nb = ncol * (long)D;
        zero_f32<<<(int)((nb + 255) / 256), 256, 0, stream>>>(agg, nb);
        scatter_edges<<<sblk, 256, 0, stream>>>(h_row, src_r2c, dst_r2c, w_r2c, agg, ne);
        gin_mlp<<<(int)(ncol / 16), 256, 0, stream>>>(
            feat_col, agg, W1_r2c, b1_r2c, W2_r2c, b2_r2c, eps_r2c, out_col, nullptr);
    }
}